// PPO_27676769255695
// MI455X (gfx1250) — compile-verified
//
#include <hip/hip_runtime.h>
#include <hip/hip_bf16.h>
#include <math.h>

#define BLOCK 256
#define IPT 8
#define EPB (BLOCK * IPT)          // 2048 elements per block
#define GAMMA_C 0.99f
#define GL_C (0.99f * 0.95f)       // gamma * gae_lambda
#define CLIP_LO 0.8f
#define CLIP_HI 1.2f

typedef __attribute__((ext_vector_type(2))) float v2f;
typedef __attribute__((ext_vector_type(8))) float v8f;

// ---------- helpers ----------

__device__ __forceinline__ void load8f(const float* p, int base, int n, float* dst) {
    if (base + IPT <= n) {
        const float4* q = reinterpret_cast<const float4*>(p + base); // base % 8 == 0 -> 32B aligned
        float4 a = q[0], b = q[1];
        dst[0]=a.x; dst[1]=a.y; dst[2]=a.z; dst[3]=a.w;
        dst[4]=b.x; dst[5]=b.y; dst[6]=b.z; dst[7]=b.w;
    } else {
        #pragma unroll
        for (int k = 0; k < IPT; ++k) dst[k] = (base + k < n) ? p[base + k] : 0.0f;
    }
}

__device__ __forceinline__ void load8b(const unsigned char* p, int base, int n, float* dst) {
    if (base + IPT <= n) {
        uint2 u = *reinterpret_cast<const uint2*>(p + base);
        #pragma unroll
        for (int k = 0; k < 4; ++k) dst[k]     = ((u.x >> (8 * k)) & 0xFFu) ? 1.0f : 0.0f;
        #pragma unroll
        for (int k = 0; k < 4; ++k) dst[4 + k] = ((u.y >> (8 * k)) & 0xFFu) ? 1.0f : 0.0f;
    } else {
        #pragma unroll
        for (int k = 0; k < IPT; ++k) dst[k] = (base + k < n && p[base + k]) ? 1.0f : 0.0f;
    }
}

// Reverse (suffix) scan of affine composites across the 256 threads of a block.
// In:  (C,D) = this thread's composite over its items (f_first ∘ ... ∘ f_last).
// Out: (iC,iD) inclusive suffix composite (threads t..255),
//      (eC,eD) exclusive suffix composite (threads t+1..255; identity for t=255).
__device__ __forceinline__ void suffix_scan(float C, float D,
                                            float* shC, float* shD,
                                            float& iC, float& iD,
                                            float& eC, float& eD) {
    const int t = threadIdx.x;
    shC[t] = C; shD[t] = D;
    __syncthreads();
    float mc = C, md = D;
    #pragma unroll
    for (int off = 1; off < BLOCK; off <<= 1) {
        float c2 = 1.0f, d2 = 0.0f;
        if (t + off < BLOCK) { c2 = shC[t + off]; d2 = shD[t + off]; }
        __syncthreads();
        md = mc * d2 + md;   // left ∘ right
        mc = mc * c2;
        shC[t] = mc; shD[t] = md;
        __syncthreads();
    }
    iC = mc; iD = md;
    eC = 1.0f; eD = 0.0f;
    if (t + 1 < BLOCK) { eC = shC[t + 1]; eD = shD[t + 1]; }
    // NOTE: caller must __syncthreads() before reusing shC/shD.
}

// Block-wide sum of one float per thread. Last 32->1 stage uses
// V_WMMA_F32_16X16X4_F32 (A = lane partials, B = ones => row sums) + shfl.
// Result valid (and uniform) on threads 0..31.
__device__ __forceinline__ float block_reduce_wmma(float v, float* sh) {
    const int t = threadIdx.x;
    __syncthreads();
    sh[t] = v;
    __syncthreads();
    float total = 0.0f;
    if (t < 32) {                     // wave 0, EXEC all ones inside this wave
        float s = 0.0f;
        #pragma unroll
        for (int k = 0; k < 8; ++k) s += sh[t + 32 * k];
        v2f a; a.x = s;    a.y = 0.0f;   // A[m][k]: lanes0-15 k=0..1, lanes16-31 k=2..3
        v2f b; b.x = 1.0f; b.y = 1.0f;   // B = all ones
        v8f c = {};
        v8f d = __builtin_amdgcn_wmma_f32_16x16x4_f32(false, a, false, b,
                                                      (short)0, c, false, false);
        float s8 = d[0]+d[1]+d[2]+d[3]+d[4]+d[5]+d[6]+d[7]; // 8 row-sums per half-wave
        total = s8 + __shfl_xor(s8, 16, 32);                 // rows 0-7 + rows 8-15
    }
    __syncthreads();
    return total;
}

__device__ __forceinline__ void make_cd(const float* rew, const float* val, const float* nv,
                                        const float* tm, const float* tr,
                                        float* c, float* d) {
    #pragma unroll
    for (int k = 0; k < IPT; ++k) {
        float term = tm[k];
        float done = (tm[k] != 0.0f || tr[k] != 0.0f) ? 1.0f : 0.0f;
        c[k] = (1.0f - done) * GL_C;
        d[k] = rew[k] + GAMMA_C * (1.0f - term) * nv[k] - val[k];
    }
}

// ---------- kernel 1: block composites + quadratic suffix statistics ----------

__global__ void __launch_bounds__(BLOCK)
ppo_k1(const float* __restrict__ rew, const float* __restrict__ val,
       const float* __restrict__ nv,
       const unsigned char* __restrict__ term, const unsigned char* __restrict__ trunc,
       int n,
       float* __restrict__ blkC, float* __restrict__ blkD,
       float* __restrict__ sumC, float* __restrict__ sumD,
       float* __restrict__ sumCC, float* __restrict__ sumCD, float* __restrict__ sumDD) {
    __shared__ float shC[BLOCK], shD[BLOCK];
    const int t = threadIdx.x;
    const int base = blockIdx.x * EPB + t * IPT;

    float r8[IPT], v8[IPT], n8[IPT], tm8[IPT], tr8[IPT], c[IPT], d[IPT];
    load8f(rew, base, n, r8); load8f(val, base, n, v8); load8f(nv, base, n, n8);
    load8b(term, base, n, tm8); load8b(trunc, base, n, tr8);
    make_cd(r8, v8, n8, tm8, tr8, c, d);

    float Ct = 1.0f, Dt = 0.0f;
    #pragma unroll
    for (int j = IPT - 1; j >= 0; --j) { Dt = c[j] * Dt + d[j]; Ct = c[j] * Ct; }

    float iC, iD, eC, eD;
    suffix_scan(Ct, Dt, shC, shD, iC, iD, eC, eD);

    // Per-element suffix composites relative to block end -> quadratic sums.
    float Cc = eC, Dc = eD;
    float aC = 0, aD = 0, aCC = 0, aCD = 0, aDD = 0;
    #pragma unroll
    for (int j = IPT - 1; j >= 0; --j) {
        Dc = c[j] * Dc + d[j];
        Cc = c[j] * Cc;
        if (base + j < n) {
            aC += Cc; aD += Dc;
            aCC += Cc * Cc; aCD += Cc * Dc; aDD += Dc * Dc;
        }
    }

    if (t == 0) { blkC[blockIdx.x] = iC; blkD[blockIdx.x] = iD; }

    float rC  = block_reduce_wmma(aC,  shC);
    float rD  = block_reduce_wmma(aD,  shC);
    float rCC = block_reduce_wmma(aCC, shC);
    float rCD = block_reduce_wmma(aCD, shC);
    float rDD = block_reduce_wmma(aDD, shC);
    if (t == 0) {
        sumC[blockIdx.x] = rC;  sumD[blockIdx.x] = rD;
        sumCC[blockIdx.x] = rCC; sumCD[blockIdx.x] = rCD; sumDD[blockIdx.x] = rDD;
    }
}

// ---------- kernel 2: scan over blocks, exact global mean / inv-std ----------

__global__ void __launch_bounds__(BLOCK)
ppo_k2(const float* __restrict__ blkC, const float* __restrict__ blkD,
       const float* __restrict__ sumC, const float* __restrict__ sumD,
       const float* __restrict__ sumCC, const float* __restrict__ sumCD,
       const float* __restrict__ sumDD,
       int B, int n,
       float* __restrict__ blkIn, float* __restrict__ stats) {
    __shared__ float shC[BLOCK], shD[BLOCK];
    __shared__ double dsum[BLOCK], dsq[BLOCK];
    const int t = threadIdx.x;
    const int SUB = (B + BLOCK - 1) / BLOCK;
    const int s0 = t * SUB;

    float Ct = 1.0f, Dt = 0.0f;
    for (int j = SUB - 1; j >= 0; --j) {
        int b = s0 + j;
        if (b < B) { Dt = blkC[b] * Dt + blkD[b]; Ct = blkC[b] * Ct; }
    }

    float iC, iD, eC, eD;
    suffix_scan(Ct, Dt, shC, shD, iC, iD, eC, eD);

    float v = eD;                       // exclusive composite applied to adv_N = 0
    double gs = 0.0, gq = 0.0;
    for (int j = SUB - 1; j >= 0; --j) {
        int b = s0 + j;
        if (b < B) {
            blkIn[b] = v;
            gs += (double)(sumC[b] * v + sumD[b]);
            gq += (double)(v * v * sumCC[b] + 2.0f * v * sumCD[b] + sumDD[b]);
            v = blkC[b] * v + blkD[b];
        }
    }

    __syncthreads();
    dsum[t] = gs; dsq[t] = gq;
    __syncthreads();
    for (int off = BLOCK / 2; off > 0; off >>= 1) {
        if (t < off) { dsum[t] += dsum[t + off]; dsq[t] += dsq[t + off]; }
        __syncthreads();
    }
    if (t == 0) {
        double S = dsum[0], Q = dsq[0];
        double mean = S / (double)n;
        double var = (Q - S * S / (double)n) / (double)(n - 1);
        if (var < 0.0) var = 0.0;
        stats[0] = (float)mean;
        stats[1] = (float)(1.0 / (sqrt(var) + 1e-9));
    }
}

// ---------- kernel 3: final elementwise outputs, LDS-staged coalesced stores ----------

__global__ void __launch_bounds__(BLOCK)
ppo_k3(const float* __restrict__ rew, const float* __restrict__ val,
       const float* __restrict__ nv,
       const float* __restrict__ lp, const float* __restrict__ olp,
       const unsigned char* __restrict__ term, const unsigned char* __restrict__ trunc,
       int n,
       const float* __restrict__ blkIn, const float* __restrict__ stats,
       float* __restrict__ out) {
    __shared__ float shC[BLOCK], shD[BLOCK];
    __shared__ float stage[3 * EPB];   // 24 KB
    const int t = threadIdx.x;
    const int base = blockIdx.x * EPB + t * IPT;

    float r8[IPT], v8[IPT], n8[IPT], lp8[IPT], op8[IPT], tm8[IPT], tr8[IPT], c[IPT], d[IPT];
    load8f(rew, base, n, r8); load8f(val, base, n, v8); load8f(nv, base, n, n8);
    load8f(lp, base, n, lp8); load8f(olp, base, n, op8);
    load8b(term, base, n, tm8); load8b(trunc, base, n, tr8);
    make_cd(r8, v8, n8, tm8, tr8, c, d);

    float Ct = 1.0f, Dt = 0.0f;
    #pragma unroll
    for (int j = IPT - 1; j >= 0; --j) { Dt = c[j] * Dt + d[j]; Ct = c[j] * Ct; }

    float iC, iD, eC, eD;
    suffix_scan(Ct, Dt, shC, shD, iC, iD, eC, eD);

    const float mean = stats[0];
    const float inv  = stats[1];
    float v = eC * blkIn[blockIdx.x] + eD;   // incoming adv at thread's last item + 1
    #pragma unroll
    for (int j = IPT - 1; j >= 0; --j) {
        v = c[j] * v + d[j];                 // adv[base + j]
        float advn  = (v - mean) * inv;
        float ret   = v + v8[j];
        float ratio = expf(lp8[j] - op8[j]);
        float cl    = fminf(fmaxf(ratio, CLIP_LO), CLIP_HI);
        float loss  = -fminf(ratio * advn, cl * advn);
        int li = t * IPT + j;
        stage[3 * li + 0] = advn;
        stage[3 * li + 1] = ret;
        stage[3 * li + 2] = loss;
    }
    __syncthreads();

    const int base3 = blockIdx.x * EPB * 3;
    const int lim3  = 3 * n;
    #pragma unroll 4
    for (int o = t; o < 3 * EPB; o += BLOCK) {
        int gi = base3 + o;
        if (gi < lim3) out[gi] = stage[o];
    }
}

// ---------- launcher ----------

extern "C" void kernel_launch(void* const* d_in, const int* in_sizes, int n_in,
                              void* d_out, int out_size, void* d_ws, size_t ws_size,
                              hipStream_t stream) {
    const int n = in_sizes[0];
    if (n <= 0) return;
    const float* rew  = (const float*)d_in[0];
    const float* val  = (const float*)d_in[1];
    const float* nv   = (const float*)d_in[2];
    const float* lp   = (const float*)d_in[3];
    const float* olp  = (const float*)d_in[4];
    const unsigned char* term  = (const unsigned char*)d_in[5];
    const unsigned char* trunc = (const unsigned char*)d_in[6];
    float* out = (float*)d_out;

    const int B = (n + EPB - 1) / EPB;
    float* ws    = (float*)d_ws;
    float* blkC  = ws + 0 * B;
    float* blkD  = ws + 1 * B;
    float* sumC  = ws + 2 * B;
    float* sumD  = ws + 3 * B;
    float* sumCC = ws + 4 * B;
    float* sumCD = ws + 5 * B;
    float* sumDD = ws + 6 * B;
    float* blkIn = ws + 7 * B;
    float* stats = ws + 8 * B;      // 2 floats

    ppo_k1<<<B, BLOCK, 0, stream>>>(rew, val, nv, term, trunc, n,
                                    blkC, blkD, sumC, sumD, sumCC, sumCD, sumDD);
    ppo_k2<<<1, BLOCK, 0, stream>>>(blkC, blkD, sumC, sumD, sumCC, sumCD, sumDD,
                                    B, n, blkIn, stats);
    ppo_k3<<<B, BLOCK, 0, stream>>>(rew, val, nv, lp, olp, term, trunc, n,
                                    blkIn, stats, out);
}